// PhysicsInformedNN_23914377904315
// MI455X (gfx1250) — compile-verified
//
#include <hip/hip_runtime.h>

// CDNA5 (gfx1250) PINN jet-propagation kernel.
// Forward value + symmetric 1st/2nd/3rd derivatives (20 jet components) of the
// 3->...->2 tanh/BN MLP, per point. Linear layers via V_WMMA_F32_16X16X4_F32
// (f32 WMMA: precision needed for 3rd derivatives), chain rule via VALU with
// branch-free hardware transcendentals (v_exp_f32 / v_rcp_f32 / v_rsq_f32).
// One wave32 owns 16 points; jet state S[comp][point][chan] lives in LDS.
// Layer geometry is specialized at compile time (templates) so the hot body
// is straight-line code the scheduler can pipeline (no per-WMMA branches).

typedef __attribute__((ext_vector_type(2))) float v2f;
typedef __attribute__((ext_vector_type(8))) float v8f;

#define BN_EPS 1e-3f
#define NCOMP 20            // 1 value + 3 first + 6 second + 10 third
#define WAVES_PER_BLOCK 2   // 2 waves * 25.6KB LDS state = 51.2KB/block

struct PinnParams {
  const float* X;
  const float* W[9];
  const float* b[8];
  const float* gamma[7];
  const float* beta[7];
  const float* mean[7];
  const float* var[7];
  const float* lambda1;
  const float* lambda2;
};

__device__ __forceinline__ v8f wmma_f32_16x16x4(v2f a, v2f b, v8f c) {
  // 8-arg form: (neg_a, A, neg_b, B, c_mod, C, reuse_a, reuse_b)
  return __builtin_amdgcn_wmma_f32_16x16x4_f32(false, a, false, b, (short)0, c,
                                               false, false);
}

// Branch-free tanh: 1 - 2/(exp(2x)+1), exp via v_exp_f32, div via v_rcp_f32.
// Saturates correctly at +-1 for large |x| (exp -> inf or 0), no EXEC churn.
__device__ __forceinline__ float fast_tanh(float x) {
  const float e = __builtin_amdgcn_exp2f(x * 2.8853900817779268f);  // exp(2x)
  return 1.0f - 2.0f * __builtin_amdgcn_rcpf(e + 1.0f);
}

// One MLP layer applied to the whole 20-component jet of 16 points.
// IND/OUTD: layer dims (compile time). NCU: jet comps with nonzero input.
template <int IND, int OUTD, int NCU>
__device__ __forceinline__ void layer_step(
    const PinnParams& P, int L, bool has_bn,
    float (&S)[WAVES_PER_BLOCK][NCOMP][16][20], int wave, int lo, int hi) {
  constexpr int NKC = (IND + 3) / 4;     // K chunks of 4
  constexpr int NNT = (OUTD + 15) / 16;  // 16-wide N tiles
  const float* Wl = P.W[L];

  // ---- stage B operands (weights) in WMMA B layout, zero-padded ----
  // B reg v, lane: K = 4*kc + 2*hi + v, N = nt*16 + lo
  // Clamped-address unconditional load + select (no EXEC-guarded loads).
  float Breg[NNT][NKC][2];
#pragma unroll
  for (int nt = 0; nt < NNT; ++nt)
#pragma unroll
    for (int kc = 0; kc < NKC; ++kc)
#pragma unroll
      for (int v = 0; v < 2; ++v) {
        const int k = 4 * kc + 2 * hi + v;
        const int n = nt * 16 + lo;
        const int kcl = (k < IND) ? k : (IND - 1);
        const int ncl = (n < OUTD) ? n : (OUTD - 1);
        const float w = Wl[kcl * OUTD + ncl];
        Breg[nt][kc][v] = (k < IND && n < OUTD) ? w : 0.0f;
      }

  // ---- Z = Jet @ W for live components, all N tiles ----
  v8f acc[NCU][NNT];
  const v8f vzero = {0, 0, 0, 0, 0, 0, 0, 0};
#pragma unroll
  for (int c = 0; c < NCU; ++c)
#pragma unroll
    for (int nt = 0; nt < NNT; ++nt) acc[c][nt] = vzero;

#pragma unroll
  for (int c = 0; c < NCU; ++c) {
#pragma unroll
    for (int kc = 0; kc < NKC; ++kc) {
      // A reg v, lane: M = lo (point), K = 4*kc + 2*hi + v  -> ds_load_b64
      v2f a = *(const v2f*)&S[wave][c][lo][4 * kc + 2 * hi];
#pragma unroll
      for (int nt = 0; nt < NNT; ++nt) {
        v2f bb;
        bb.x = Breg[nt][kc][0];
        bb.y = Breg[nt][kc][1];
        acc[c][nt] = wmma_f32_16x16x4(a, bb, acc[c][nt]);
      }
    }
  }

  // ---- per-element tanh chain rule (3rd order) + BN, store back ----
  // D layout: reg r, lane -> point = r + 8*hi, channel = nt*16 + lo
  constexpr int PI[3][3] = {{4, 5, 6}, {5, 7, 8}, {6, 8, 9}};
  constexpr int T3[10][3] = {{0, 0, 0}, {0, 0, 1}, {0, 0, 2}, {0, 1, 1},
                             {0, 1, 2}, {0, 2, 2}, {1, 1, 1}, {1, 1, 2},
                             {1, 2, 2}, {2, 2, 2}};
#pragma unroll
  for (int nt = 0; nt < NNT; ++nt) {
    const int ch = nt * 16 + lo;
    const bool chok = ch < OUTD;
    const int chc = chok ? ch : (OUTD - 1);   // clamped, always-valid index
    float bias = P.b[L][chc];
    bias = chok ? bias : 0.0f;
    float bn_s = 1.0f, bn_m = 0.0f, bn_b = 0.0f;
    if (has_bn) {  // wave-uniform
      const float g  = P.gamma[L][chc];
      const float vr = P.var[L][chc];
      const float mn = P.mean[L][chc];
      const float bt = P.beta[L][chc];
      bn_s = chok ? g * __builtin_amdgcn_rsqf(vr + BN_EPS) : 0.0f;
      bn_m = chok ? mn : 0.0f;
      bn_b = chok ? bt : 0.0f;
    }
#pragma unroll
    for (int r = 0; r < 8; ++r) {
      float z[NCOMP], h[NCOMP];
#pragma unroll
      for (int c = 0; c < NCOMP; ++c)
        z[c] = (c < NCU) ? acc[c < NCU ? c : 0][nt][r] : 0.0f;
      z[0] += bias;
      const float tv = fast_tanh(z[0]);
      const float g1 = 1.0f - tv * tv;                // tanh'
      const float g2 = -2.0f * tv * g1;               // tanh''
      const float g3 = g1 * (6.0f * tv * tv - 2.0f);  // tanh'''
      h[0] = tv;
      h[1] = g1 * z[1];
      h[2] = g1 * z[2];
      h[3] = g1 * z[3];
      h[4] = g2 * z[1] * z[1] + g1 * z[4];
      h[5] = g2 * z[1] * z[2] + g1 * z[5];
      h[6] = g2 * z[1] * z[3] + g1 * z[6];
      h[7] = g2 * z[2] * z[2] + g1 * z[7];
      h[8] = g2 * z[2] * z[3] + g1 * z[8];
      h[9] = g2 * z[3] * z[3] + g1 * z[9];
#pragma unroll
      for (int m = 0; m < 10; ++m) {
        const int a = T3[m][0], bq = T3[m][1], cq = T3[m][2];
        h[10 + m] = g3 * z[1 + a] * z[1 + bq] * z[1 + cq] +
                    g2 * (z[PI[a][bq]] * z[1 + cq] +
                          z[PI[a][cq]] * z[1 + bq] +
                          z[PI[bq][cq]] * z[1 + a]) +
                    g1 * z[10 + m];
      }
      if (has_bn) {
        h[0] = (h[0] - bn_m) * bn_s + bn_b;
#pragma unroll
        for (int c = 1; c < NCOMP; ++c) h[c] *= bn_s;
      }
      if (chok) {  // EXEC-masked stores only (outside WMMA region)
        const int p = r + 8 * hi;
#pragma unroll
        for (int c = 0; c < NCOMP; ++c) S[wave][c][p][ch] = h[c];
      }
    }
  }
  asm volatile("s_wait_dscnt 0" ::: "memory");
}

__global__ __launch_bounds__(32 * WAVES_PER_BLOCK)
void pinn_jet_kernel(PinnParams P, float* __restrict__ out, int N) {
  const int wave = threadIdx.x >> 5;
  const int lane = threadIdx.x & 31;
  const int lo   = lane & 15;   // A: M row / B,C,D: N column
  const int hi   = lane >> 4;   // A,B: K half-select / C,D: M +8
  const int gp0  = (blockIdx.x * WAVES_PER_BLOCK + wave) * 16;

  // Per-wave jet state: [comp][point][chan], chan stride 20 (8B aligned rows).
  __shared__ float S[WAVES_PER_BLOCK][NCOMP][16][20];

  if (gp0 >= N) return;

  // ---- zero state, then seed value + unit first-derivative directions ----
  float* Sw = &S[wave][0][0][0];
  for (int i = lane; i < NCOMP * 16 * 20; i += 32) Sw[i] = 0.0f;
  if (lane < 16) {
    const int gp = gp0 + lane;
    S[wave][0][lane][0] = P.X[3 * gp + 0];
    S[wave][0][lane][1] = P.X[3 * gp + 1];
    S[wave][0][lane][2] = P.X[3 * gp + 2];
    S[wave][1][lane][0] = 1.0f;   // d/dx
    S[wave][2][lane][1] = 1.0f;   // d/dy
    S[wave][3][lane][2] = 1.0f;   // d/dt
  }
  asm volatile("s_wait_dscnt 0" ::: "memory");

  // Layer 0 (3->3): only value + 3 first-derivative comps are nonzero.
  layer_step<3, 3, 4>(P, 0, true, S, wave, lo, hi);
  // Layer 1 (3->20): full jet is live after layer 0's chain rule.
  layer_step<3, 20, 20>(P, 1, true, S, wave, lo, hi);
  // Layers 2..7 (20->20): single specialized body, runtime BN flag.
  for (int L = 2; L < 8; ++L)
    layer_step<20, 20, 20>(P, L, (L < 7), S, wave, lo, hi);

  // ---- final linear layer W8: 20 -> 2 (psi, p), then NS residuals ----
  float w80[20], w81[20];
#pragma unroll
  for (int i = 0; i < 20; ++i) {
    w80[i] = P.W[8][2 * i + 0];
    w81[i] = P.W[8][2 * i + 1];
  }
  const float l1 = P.lambda1[0];
  const float l2 = P.lambda2[0];
  const int p = lo;  // hi lanes duplicate; only lanes<16 store
  auto dot = [&](int c, const float* wv) {
    float s = 0.0f;
#pragma unroll
    for (int i = 0; i < 20; ++i) s += S[wave][c][p][i] * wv[i];
    return s;
  };
  // comp index map: 0 val; 1..3 d/dx,dy,dt; 4..9 (xx,xy,xt,yy,yt,tt);
  // 10..19 (xxx,xxy,xxt,xyy,xyt,xtt,yyy,yyt,ytt,ttt)
  const float u    =  dot(2, w80);    // psi_y
  const float v    = -dot(1, w80);    // -psi_x
  const float pr   =  dot(0, w81);    // p
  const float p_x  =  dot(1, w81);
  const float p_y  =  dot(2, w81);
  const float u_x  =  dot(5, w80);    // psi_xy
  const float u_y  =  dot(7, w80);    // psi_yy
  const float u_t  =  dot(8, w80);    // psi_yt
  const float v_x  = -dot(4, w80);    // -psi_xx
  const float v_y  = -dot(5, w80);    // -psi_xy
  const float v_t  = -dot(6, w80);    // -psi_xt
  const float u_xx =  dot(11, w80);   // psi_xxy
  const float u_yy =  dot(16, w80);   // psi_yyy
  const float v_xx = -dot(10, w80);   // -psi_xxx
  const float v_yy = -dot(13, w80);   // -psi_xyy
  const float f_u = u_t + l1 * (u * u_x + v * u_y) + p_x - l2 * (u_xx + u_yy);
  const float f_v = v_t + l1 * (u * v_x + v * v_y) + p_y - l2 * (v_xx + v_yy);

  if (lane < 16) {
    const int gp = gp0 + p;
    out[0 * N + gp] = u;
    out[1 * N + gp] = v;
    out[2 * N + gp] = pr;
    out[3 * N + gp] = f_u;
    out[4 * N + gp] = f_v;
  }
}

extern "C" void kernel_launch(void* const* d_in, const int* in_sizes, int n_in,
                              void* d_out, int out_size, void* d_ws,
                              size_t ws_size, hipStream_t stream) {
  (void)n_in; (void)out_size; (void)d_ws; (void)ws_size;
  PinnParams P;
  P.X = (const float*)d_in[0];
  for (int i = 0; i < 9; ++i) P.W[i]     = (const float*)d_in[1 + i];
  for (int i = 0; i < 8; ++i) P.b[i]     = (const float*)d_in[10 + i];
  for (int i = 0; i < 7; ++i) P.gamma[i] = (const float*)d_in[18 + i];
  for (int i = 0; i < 7; ++i) P.beta[i]  = (const float*)d_in[25 + i];
  for (int i = 0; i < 7; ++i) P.mean[i]  = (const float*)d_in[32 + i];
  for (int i = 0; i < 7; ++i) P.var[i]   = (const float*)d_in[39 + i];
  P.lambda1 = (const float*)d_in[46];
  P.lambda2 = (const float*)d_in[47];

  const int N = in_sizes[0] / 3;                 // 65536 points
  const int pts_per_block = 16 * WAVES_PER_BLOCK;
  const int blocks = (N + pts_per_block - 1) / pts_per_block;
  pinn_jet_kernel<<<blocks, 32 * WAVES_PER_BLOCK, 0, stream>>>(
      P, (float*)d_out, N);
}